// KVCache_71966472012644
// MI455X (gfx1250) — compile-verified
//
#include <hip/hip_runtime.h>
#include <hip/hip_bf16.h>
#include <stdint.h>

// ---------------------------------------------------------------------------
// Paged KV-cache append (flashinfer append_paged_kv_cache semantics).
// Strategy: build inverse page map in d_ws, then ONE output-linear pass over
// the 512 MiB cache, sourcing each 16B chunk from k/v (appended slots) or the
// input cache (untouched slots). Data movement is double-buffered through LDS
// with the CDNA5 async global<->LDS DMA path (ASYNCcnt), one wait per 2KB.
// ---------------------------------------------------------------------------

#define GLOBAL_AS __attribute__((address_space(1)))
#define LDS_AS    __attribute__((address_space(3)))

typedef int v4i_vec __attribute__((vector_size(16)));   // matches builtin param type
typedef GLOBAL_AS v4i_vec* gptr_b128;
typedef LDS_AS    v4i_vec* lptr_b128;

#if defined(__gfx1250__) &&                                              \
    __has_builtin(__builtin_amdgcn_global_load_async_to_lds_b128) &&     \
    __has_builtin(__builtin_amdgcn_global_store_async_from_lds_b128)
#define USE_ASYNC_LDS 1
#else
#define USE_ASYNC_LDS 0
#endif

static constexpr int PAGE_SIZE = 16;
static constexpr int N_HEADS   = 8;
static constexpr int HEAD_DIM  = 128;
static constexpr int ROW_V4    = (N_HEADS * HEAD_DIM) / 8;   // 128 uint4 per (page,kv,slot) row
static constexpr int PAGE_V4   = 2 * PAGE_SIZE * ROW_V4;     // 4096 uint4 per page
static constexpr int EMIT_BLOCK  = 256;                      // 8 wave32
static constexpr int EMIT_UNROLL = 4;                        // 4 x 512B per wave per chunk

// --------------------------- inverse page map ------------------------------

__global__ void inv_init_kernel(int* __restrict__ inv, int n) {
  int i = blockIdx.x * blockDim.x + threadIdx.x;
  if (i < n) inv[i] = -1;
}

__global__ void inv_build_kernel(const int* __restrict__ page_indices,
                                 const int* __restrict__ page_indptr,
                                 int B, int n, int* __restrict__ inv) {
  int j = blockIdx.x * blockDim.x + threadIdx.x;
  if (j >= n) return;
  int b = 0;
  while (b + 1 < B && page_indptr[b + 1] <= j) ++b;   // CSR searchsorted (B tiny)
  int pageofs = j - page_indptr[b];
  inv[page_indices[j]] = (b << 20) | pageofs;
}

// --------------------------- source selection ------------------------------

__device__ __forceinline__ const uint4* src_for(
    uint32_t i, const uint4* __restrict__ cache_in,
    const uint4* __restrict__ k, const uint4* __restrict__ v,
    const int* __restrict__ inv,
    const int* __restrict__ append_indptr,
    const int* __restrict__ page_indptr,
    const int* __restrict__ lastlen) {
  uint32_t page = i >> 12;            // i / PAGE_V4
  uint32_t r    = i & 4095u;
  uint32_t kv   = r >> 11;            // 0 = K half, 1 = V half
  uint32_t slot = (r >> 7) & 15u;
  uint32_t off  = r & 127u;           // uint4 within 2KB row
  int m = inv[page];
  if (m >= 0) {
    int b       = m >> 20;
    int pageofs = m & 0xFFFFF;
    int npages  = page_indptr[b + 1] - page_indptr[b];
    int kv_len  = (npages - 1) * PAGE_SIZE + lastlen[b];
    int app_beg = append_indptr[b];
    int app_len = append_indptr[b + 1] - app_beg;
    int pos     = pageofs * PAGE_SIZE + (int)slot;
    int tokidx  = pos - (kv_len - app_len);
    if (tokidx >= 0 && tokidx < app_len) {
      int token = app_beg + tokidx;
      const uint4* base = kv ? v : k;
      return base + (size_t)token * ROW_V4 + off;
    }
  }
  return cache_in + i;                // untouched slot: pass input cache through
}

// ------------------------------ emit pass ----------------------------------

__global__ __launch_bounds__(EMIT_BLOCK)
void emit_kernel(const uint4* __restrict__ cache_in,
                 const uint4* __restrict__ k, const uint4* __restrict__ v,
                 const int* __restrict__ inv,
                 const int* __restrict__ append_indptr,
                 const int* __restrict__ page_indptr,
                 const int* __restrict__ lastlen,
                 uint32_t nvec, uint4* __restrict__ out) {
#if USE_ASYNC_LDS
  // 8 waves x 2 slabs x (4 x 512B); 16B-aligned by type. 32 KB / block.
  __shared__ uint4 smem[(EMIT_BLOCK / 32) * 2 * EMIT_UNROLL * 32];

  const uint32_t lane  = threadIdx.x & 31u;
  const uint32_t wib   = threadIdx.x >> 5;
  const uint32_t w     = blockIdx.x * (blockDim.x >> 5) + wib;
  const uint32_t nw    = gridDim.x * (blockDim.x >> 5);
  const uint32_t CHUNK = EMIT_UNROLL * 32;                 // 128 uint4 / wave / chunk
  const uint32_t nvec_main = nvec & ~(CHUNK - 1u);
  uint4* const slab0 = &smem[(wib * 2 + 0) * CHUNK];
  uint4* const slab1 = &smem[(wib * 2 + 1) * CHUNK];

  auto issue_loads = [&](uint32_t base, uint4* slab) {
#pragma unroll
    for (int u = 0; u < EMIT_UNROLL; ++u) {
      uint32_t i = base + u * 32 + lane;
      const uint4* s =
          src_for(i, cache_in, k, v, inv, append_indptr, page_indptr, lastlen);
      __builtin_amdgcn_global_load_async_to_lds_b128(
          (gptr_b128)s, (lptr_b128)&slab[u * 32 + lane], 0, 0);
    }
  };
  auto issue_stores = [&](uint32_t base, const uint4* slab) {
#pragma unroll
    for (int u = 0; u < EMIT_UNROLL; ++u) {
      uint32_t i = base + u * 32 + lane;
      __builtin_amdgcn_global_store_async_from_lds_b128(
          (gptr_b128)(out + i), (lptr_b128)&slab[u * 32 + lane], 0, 0);
    }
  };
  auto wait_async0 = [&]() {
#if __has_builtin(__builtin_amdgcn_s_wait_asynccnt)
    __builtin_amdgcn_s_wait_asynccnt(0);
#else
    asm volatile("s_wait_asynccnt 0" ::: "memory");
#endif
  };

  uint32_t base = w * CHUNK;
  if (base < nvec_main) {
    uint4* cur = slab0;
    uint4* nxt = slab1;
    issue_loads(base, cur);                       // prologue: fill slab0
    while (base < nvec_main) {
      uint32_t next = base + nw * CHUNK;
      wait_async0();            // cur's loads done (and prior stores drained)
      issue_stores(base, cur);  // drain cur -> out
      if (next < nvec_main)
        issue_loads(next, nxt); // refill other slab; overlaps with stores
      uint4* t = cur; cur = nxt; nxt = t;
      base = next;
    }
    // final stores drain under s_endpgm's implicit wait-idle
  }
  // tail (empty for this problem size; kept for generality)
  uint32_t tid = blockIdx.x * blockDim.x + threadIdx.x;
  for (uint32_t i = nvec_main + tid; i < nvec; i += gridDim.x * blockDim.x)
    out[i] = *src_for(i, cache_in, k, v, inv, append_indptr, page_indptr, lastlen);
#else
  // Plain b128 fallback (also used for the host pass).
  uint32_t tid    = blockIdx.x * blockDim.x + threadIdx.x;
  uint32_t stride = gridDim.x * blockDim.x;
  for (uint32_t i = tid; i < nvec; i += stride)
    out[i] = *src_for(i, cache_in, k, v, inv, append_indptr, page_indptr, lastlen);
#endif
}

// ------------------------------ launcher -----------------------------------

extern "C" void kernel_launch(void* const* d_in, const int* in_sizes, int n_in,
                              void* d_out, int out_size, void* d_ws, size_t ws_size,
                              hipStream_t stream) {
  const __hip_bfloat16* k        = (const __hip_bfloat16*)d_in[0];
  const __hip_bfloat16* v        = (const __hip_bfloat16*)d_in[1];
  const __hip_bfloat16* cache_in = (const __hip_bfloat16*)d_in[2];
  const int* append_indptr       = (const int*)d_in[3];
  const int* page_indices        = (const int*)d_in[4];
  const int* page_indptr         = (const int*)d_in[5];
  const int* lastlen             = (const int*)d_in[6];

  const int B          = in_sizes[3] - 1;
  const int n_page_ent = in_sizes[4];
  const int max_pages  = in_sizes[2] / (2 * PAGE_SIZE * N_HEADS * HEAD_DIM);
  const uint32_t nvec  = (uint32_t)(out_size / 8);   // uint4 = 8 bf16

  int* inv = (int*)d_ws;  // max_pages * 4 bytes

  inv_init_kernel<<<(max_pages + 255) / 256, 256, 0, stream>>>(inv, max_pages);
  inv_build_kernel<<<(n_page_ent + 255) / 256, 256, 0, stream>>>(
      page_indices, page_indptr, B, n_page_ent, inv);
  emit_kernel<<<2048, EMIT_BLOCK, 0, stream>>>(
      (const uint4*)cache_in, (const uint4*)k, (const uint4*)v, inv,
      append_indptr, page_indptr, lastlen, nvec, (uint4*)d_out);
}